// DDSP_23545010717501
// MI455X (gfx1250) — compile-verified
//
#include <hip/hip_runtime.h>
#include <math.h>

#define Bn   16
#define Tn   400
#define NHn  100
#define NBn  65
#define SRn  16000
#define BLKn 160
#define Nn   (Tn * BLKn)   /* 64000 */
#define RLn  16000
#define PI_F 3.14159265358979323846f

typedef __attribute__((ext_vector_type(2))) float v2f;
typedef __attribute__((ext_vector_type(8))) float v8f;

// ---------------- Kernel 1: per-batch exclusive scan of per-block phase steps ----------------
// omega[b, t*160+i] = phase0[b,t] + (i+1)*delta[b,t]; phase0 = exclusive prefix of 160*delta.
__global__ void k_scan(const float* __restrict__ pitch, float* __restrict__ phase0,
                       float* __restrict__ delta) {
  __shared__ float sc[512];
  int b = blockIdx.x, t = threadIdx.x;
  float dl = 0.f, bs = 0.f;
  if (t < Tn) {
    float p = pitch[b * Tn + t];
    dl = 2.f * PI_F * p / (float)SRn;
    bs = (float)BLKn * dl;
  }
  sc[t] = bs;
  __syncthreads();
  for (int off = 1; off < 512; off <<= 1) {
    float v = (t >= off) ? sc[t - off] : 0.f;
    __syncthreads();
    sc[t] += v;
    __syncthreads();
  }
  if (t < Tn) {
    phase0[b * Tn + t] = sc[t] - bs;   // exclusive prefix
    delta[b * Tn + t]  = dl;
  }
}

// ---------------- Kernel 2: anti-aliased, normalized, scaled harmonic amplitudes ----------------
__global__ void k_amps(const float* __restrict__ pitch, const float* __restrict__ total_amp,
                       const float* __restrict__ ha, float* __restrict__ amps) {
  __shared__ float red[128];
  int bt = blockIdx.x, k = threadIdx.x;
  int b = bt / Tn, t = bt % Tn;
  float p = pitch[bt];
  float v = 0.f;
  if (k < NHn) {
    float f  = p * (float)(k + 1);
    float aa = ((f < 0.5f * (float)SRn) ? 1.f : 0.f) + 1e-4f;
    v = ha[(size_t)(b * NHn + k) * Tn + t] * aa;
  }
  red[k] = v;
  __syncthreads();
  for (int off = 64; off > 0; off >>= 1) {
    if (k < off) red[k] += red[k + off];
    __syncthreads();
  }
  float scale = total_amp[bt] / red[0];
  if (k < NHn) amps[(size_t)bt * NHn + k] = v * scale;
}

// ---------------- Kernel 3: windowed-cosine irfft matrix (B-operand of the GEMM) ----------------
// impulse[j] = sum_k Bmat[k][j] * amp[k]; rows k>=65 zero-padded (K padded to 68).
__global__ void k_table(float* __restrict__ Bmat) {
  for (int idx = threadIdx.x; idx < 68 * BLKn; idx += blockDim.x) {
    int kk = idx / BLKn, n = idx % BLKn;
    float val = 0.f;
    if (kk < NBn && !(n >= 64 && n < 96)) {
      int m    = (n < 64) ? (n + 64) : (n - 96);   // undo roll(-64) after pad
      int nIdx = (m + 64) & 127;                   // undo roll(+64) of the irfft
      float wgt   = 0.5f - 0.5f * cosf(2.f * PI_F * (float)m / 128.f); // periodic Hann
      float scale = ((kk == 0) || (kk == 64)) ? (1.f / 128.f) : (2.f / 128.f);
      int kn = (kk * nIdx) & 127;                  // cosine arg mod 2*pi
      val = wgt * scale * cosf(2.f * PI_F * (float)kn / 128.f);
    }
    Bmat[idx] = val;
  }
}

// ---------------- Kernel 4: impulse = noise_filter[6400x65] @ Bmat[65x160] via WMMA f32 ----------
// One block = one 16-row M tile; 10 waves each own one 16-col N tile; 17 K-steps of 4.
__global__ void k_ir_gemm(const float* __restrict__ nf, const float* __restrict__ Bmat,
                          float* __restrict__ impulse) {
  int lane = threadIdx.x & 31;
  int wave = threadIdx.x >> 5;
  int row0 = blockIdx.x * 16;
  int n0   = wave * 16;
  int mi   = lane & 15;
  int hi   = lane >> 4;                      // half-wave select per ISA 7.12.2
  const float* Arow = nf + (size_t)(row0 + mi) * NBn;
  v8f acc = {};
#pragma unroll
  for (int kk = 0; kk < 17; ++kk) {
    int ka0 = kk * 4 + hi * 2;               // A VGPR0: K=0 (lo half) / K=2 (hi half)
    int ka1 = ka0 + 1;                       // A VGPR1: K=1 / K=3
    v2f a, b;
    a.x = (ka0 < NBn) ? Arow[ka0] : 0.f;
    a.y = (ka1 < NBn) ? Arow[ka1] : 0.f;
    b.x = Bmat[ka0 * BLKn + n0 + mi];        // rows 65..67 are zero-padded
    b.y = Bmat[ka1 * BLKn + n0 + mi];
    acc = __builtin_amdgcn_wmma_f32_16x16x4_f32(
        /*neg_a=*/false, a, /*neg_b=*/false, b,
        /*c_mod=*/(short)0, acc, /*reuse_a=*/false, /*reuse_b=*/false);
  }
  float* orow = impulse + (size_t)row0 * BLKn;
#pragma unroll
  for (int v = 0; v < 8; ++v) {              // C/D: VGPR v -> M=v (lo half) / v+8 (hi half)
    int M = v + hi * 8;
    orow[(size_t)M * BLKn + n0 + mi] = acc[v];
  }
}

// ---------------- Kernel 5: harmonic synthesis via Chebyshev sin recurrence ----------------
__global__ void k_harm(const float* __restrict__ amps, const float* __restrict__ phase0,
                       const float* __restrict__ delta, float* __restrict__ signal) {
  __shared__ float amp_s[NHn];
  int bt = blockIdx.x, i = threadIdx.x;
  int b = bt / Tn, t = bt % Tn;
  if (i < NHn) amp_s[i] = amps[(size_t)bt * NHn + i];
  __syncthreads();
  float w = phase0[bt] + (float)(i + 1) * delta[bt];
  float s, c;
  sincosf(w, &s, &c);
  float two_c = 2.f * c;
  float skm1 = 0.f, sk = s, acc = 0.f;
#pragma unroll 4
  for (int k = 0; k < NHn; ++k) {
    acc = fmaf(amp_s[k], sk, acc);
    float nx = fmaf(two_c, sk, -skm1);       // sin((k+2)w) = 2cos(w)sin((k+1)w) - sin(kw)
    skm1 = sk; sk = nx;
  }
  signal[(size_t)b * Nn + (size_t)t * BLKn + i] = acc;
}

// ---------------- Kernel 6: 160-tap causal FIR (the fft_convolve tail) + add to harmonic --------
__global__ void k_noise(const float* __restrict__ noise, const float* __restrict__ impulse,
                        float* __restrict__ signal) {
  __shared__ float nz[BLKn], im[BLKn];
  int bt = blockIdx.x, j = threadIdx.x;
  int b = bt / Tn, t = bt % Tn;
  nz[j] = noise[(size_t)bt * BLKn + j];
  im[j] = impulse[(size_t)bt * BLKn + j];
  __syncthreads();
  float acc = 0.f;
  for (int m = 0; m <= j; ++m) acc = fmaf(nz[m], im[j - m], acc);
  size_t idx = (size_t)b * Nn + (size_t)t * BLKn + j;
  signal[idx] += acc;
}

// ---------------- Kernel 7: reverb impulse (decayed noise, element 0 forced to 1.0) -------------
__global__ void k_rimp(const float* __restrict__ rn, const float* __restrict__ dec,
                       const float* __restrict__ wet, float* __restrict__ rimp) {
  int l = blockIdx.x * blockDim.x + threadIdx.x;
  if (l >= RLn) return;
  float d = dec[0], w = wet[0];
  float sp = log1pf(expf(-d));               // softplus(-decay)
  float sg = 1.f / (1.f + expf(-w));         // sigmoid(wet)
  float tsec = (float)l / (float)SRn;
  float v = rn[l] * expf(-sp * tsec * 500.f) * sg;
  rimp[l] = (l == 0) ? 1.f : v;
}

// ---------------- Kernel 8: 16000-tap causal reverb FIR ----------------
// Double-buffered in LDS via CDNA5 async global->LDS loads (ASYNCcnt), scalar-uniform taps.
__global__ void k_reverb(const float* __restrict__ signal, const float* __restrict__ rimp,
                         float* __restrict__ out) {
  const int CH = 128;
  const int W  = CH + 256;                   // 384-float signal window
  __shared__ float simp[2][W];
  int b   = blockIdx.y;
  int j0  = blockIdx.x * 256;
  int tid = threadIdx.x;
  const float* sb = signal + (size_t)b * Nn;
  float acc = 0.f;
  int Dmax    = (RLn < j0 + 256) ? RLn : (j0 + 256);  // always a multiple of CH
  int nChunks = Dmax / CH;

  // Issue async fill of buffer `buf` with the window for tap-chunk base D.
  // Lane i covers simp[buf][i]; lanes < CH also cover simp[buf][256+i].
  auto issue = [&](int buf, int D) {
    int base = j0 - D - (CH - 1);
    {
      int gi = base + tid;
      unsigned lo = (unsigned)(unsigned long long)&simp[buf][tid];
      const float* ga = sb + (gi < 0 ? 0 : gi);        // clamp; zero-patched after wait
      asm volatile("global_load_async_to_lds_b32 %0, %1, off"
                   :: "v"(lo), "v"(ga) : "memory");
    }
    if (tid < CH) {
      int gi = base + 256 + tid;
      unsigned lo = (unsigned)(unsigned long long)&simp[buf][256 + tid];
      const float* ga = sb + (gi < 0 ? 0 : gi);
      asm volatile("global_load_async_to_lds_b32 %0, %1, off"
                   :: "v"(lo), "v"(ga) : "memory");
    }
  };
  // Zero-patch slots whose source index was negative (j-d < 0 region).
  auto fix = [&](int buf, int D) {
    int base = j0 - D - (CH - 1);
    if (base + tid < 0) simp[buf][tid] = 0.f;
    if (tid < CH && base + 256 + tid < 0) simp[buf][256 + tid] = 0.f;
  };

  // Prologue: bring in chunk 0.
  issue(0, 0);
  asm volatile("s_wait_asynccnt 0x0" ::: "memory");
  fix(0, 0);
  __syncthreads();

  for (int c = 0; c < nChunks; ++c) {
    int D = c * CH;
    int cur = c & 1;
    if (c + 1 < nChunks) issue(cur ^ 1, D + CH);       // overlap next fill with compute
    const float* sc = simp[cur];
#pragma unroll 8
    for (int dd = 0; dd < CH; ++dd)                    // taps are wave-uniform -> SMEM
      acc = fmaf(rimp[D + dd], sc[tid + (CH - 1) - dd], acc);
    asm volatile("s_wait_asynccnt 0x0" ::: "memory");  // own transfers done before barrier
    if (c + 1 < nChunks) fix(cur ^ 1, D + CH);
    __syncthreads();                                   // all waves' fills now visible
  }
  out[(size_t)b * Nn + j0 + tid] = acc;
}

extern "C" void kernel_launch(void* const* d_in, const int* in_sizes, int n_in,
                              void* d_out, int out_size, void* d_ws, size_t ws_size,
                              hipStream_t stream) {
  (void)in_sizes; (void)n_in; (void)out_size; (void)ws_size;
  const float* pitch     = (const float*)d_in[0];
  const float* total_amp = (const float*)d_in[1];
  const float* harmo     = (const float*)d_in[2];
  const float* nf        = (const float*)d_in[3];
  const float* noise     = (const float*)d_in[4];
  const float* rnoise    = (const float*)d_in[5];
  const float* decay     = (const float*)d_in[6];
  const float* wet       = (const float*)d_in[7];

  float* ws      = (float*)d_ws;
  float* amps    = ws;                                // 640000
  float* phase0  = amps    + (size_t)Bn * Tn * NHn;   // 6400
  float* delta   = phase0  + (size_t)Bn * Tn;         // 6400
  float* Bmat    = delta   + (size_t)Bn * Tn;         // 68*160
  float* impulse = Bmat    + (size_t)68 * BLKn;       // 6400*160
  float* signal  = impulse + (size_t)Bn * Tn * BLKn;  // 16*64000
  float* rimp    = signal  + (size_t)Bn * Nn;         // 16000
  float* out     = (float*)d_out;

  k_scan   <<<Bn, 512, 0, stream>>>(pitch, phase0, delta);
  k_amps   <<<Bn * Tn, 128, 0, stream>>>(pitch, total_amp, harmo, amps);
  k_table  <<<1, 256, 0, stream>>>(Bmat);
  k_ir_gemm<<<(Bn * Tn) / 16, 320, 0, stream>>>(nf, Bmat, impulse);
  k_harm   <<<Bn * Tn, BLKn, 0, stream>>>(amps, phase0, delta, signal);
  k_noise  <<<Bn * Tn, BLKn, 0, stream>>>(noise, impulse, signal);
  k_rimp   <<<(RLn + 255) / 256, 256, 0, stream>>>(rnoise, decay, wet, rimp);
  k_reverb <<<dim3(Nn / 256, Bn), 256, 0, stream>>>(signal, rimp, out);
}